// JaccardLoss_10900626997865
// MI455X (gfx1250) — compile-verified
//
#include <hip/hip_runtime.h>
#include <stdint.h>

// Problem constants (from the reference)
#define BATCH 64
#define NELEM 262144                 // 512*512 per row, power of two
#define BPR   32                     // blocks per row
#define CHUNK (NELEM / BPR)          // 8192 floats per block per array
#define TILE  2048                   // floats per array per tile (8 KB)
#define NTILES (CHUNK / TILE)        // 4
#define TPB   256                    // 8 wave32s
#define V4_PER_THREAD (TILE / TPB / 4) // 2 float4 per thread per tile
#define PART_STRIDE 8                // 7 floats + pad

struct Red {
    float sp, st, spt, n1;   // sums: p, t, p*t, count(p>0.5)
    float mx, ce, ces;       // running max(t), count(t==mx), count(t==mx && p>0.5)
};

__device__ __forceinline__ void combine(Red& a, float omx, float oce, float oces) {
    if (omx > a.mx)      { a.mx = omx; a.ce = oce; a.ces = oces; }
    else if (omx == a.mx){ a.ce += oce; a.ces += oces; }
}

__global__ __launch_bounds__(TPB)
void jaccard_partial(const float* __restrict__ probs,
                     const float* __restrict__ targets,
                     float* __restrict__ part) {
    __shared__ float lp[2][TILE];
    __shared__ float lt[2][TILE];
    __shared__ float red[TPB / 32][8];

    const int row = blockIdx.y;
    const int blk = blockIdx.x;
    const int tid = threadIdx.x;
    const size_t base = (size_t)row * NELEM + (size_t)blk * CHUNK;
    const float* gp = probs + base;
    const float* gt = targets + base;

    // Issue one tile's worth of async global->LDS copies.
    // Per wave: V4_PER_THREAD*2 = 4 async instructions -> ASYNCcnt += 4.
    auto issue_tile = [&](int t, int buf) {
        uint32_t lpOff = (uint32_t)(uintptr_t)(&lp[buf][0]);
        uint32_t ltOff = (uint32_t)(uintptr_t)(&lt[buf][0]);
#pragma unroll
        for (int j = 0; j < V4_PER_THREAD; ++j) {
            uint32_t e = (uint32_t)(j * TPB + tid) * 4u;           // float index in tile
            uint64_t gaP = (uint64_t)(uintptr_t)(gp + (size_t)t * TILE + e);
            uint64_t gaT = (uint64_t)(uintptr_t)(gt + (size_t)t * TILE + e);
            uint32_t laP = lpOff + e * 4u;
            uint32_t laT = ltOff + e * 4u;
            asm volatile("global_load_async_to_lds_b128 %0, %1, off"
                         :: "v"(laP), "v"(gaP) : "memory");
            asm volatile("global_load_async_to_lds_b128 %0, %1, off"
                         :: "v"(laT), "v"(gaT) : "memory");
        }
    };

    Red r;
    r.sp = 0.f; r.st = 0.f; r.spt = 0.f; r.n1 = 0.f;
    r.mx = -__builtin_inff(); r.ce = 0.f; r.ces = 0.f;

    auto acc1 = [&](float x, float y) {
        r.sp += x; r.st += y; r.spt += x * y;
        float sr = (x > 0.5f) ? 1.0f : 0.0f;
        r.n1 += sr;
        if (y > r.mx)      { r.mx = y; r.ce = 1.0f; r.ces = sr; }
        else if (y == r.mx){ r.ce += 1.0f; r.ces += sr; }
    };

    issue_tile(0, 0);
    for (int t = 0; t < NTILES; ++t) {
        const int buf = t & 1;
        if (t + 1 < NTILES) {
            issue_tile(t + 1, buf ^ 1);
            // 4 ops of the NEXT tile may stay outstanding; current tile's 4 must land.
            asm volatile("s_wait_asynccnt 4" ::: "memory");
        } else {
            asm volatile("s_wait_asynccnt 0" ::: "memory");
        }
        __syncthreads();   // whole tile visible to all waves

#pragma unroll
        for (int j = 0; j < V4_PER_THREAD; ++j) {
            int e = (j * TPB + tid) * 4;
            float4 p4 = *reinterpret_cast<const float4*>(&lp[buf][e]);
            float4 t4 = *reinterpret_cast<const float4*>(&lt[buf][e]);
            acc1(p4.x, t4.x); acc1(p4.y, t4.y);
            acc1(p4.z, t4.z); acc1(p4.w, t4.w);
        }
        __syncthreads();   // all waves done reading before buffer is refilled
    }

    // wave32 shuffle reduction (lane 0 correct)
    for (int off = 16; off > 0; off >>= 1) {
        r.sp  += __shfl_down(r.sp,  off);
        r.st  += __shfl_down(r.st,  off);
        r.spt += __shfl_down(r.spt, off);
        r.n1  += __shfl_down(r.n1,  off);
        float omx  = __shfl_down(r.mx,  off);
        float oce  = __shfl_down(r.ce,  off);
        float oces = __shfl_down(r.ces, off);
        combine(r, omx, oce, oces);
    }

    const int lane = tid & 31;
    const int wv   = tid >> 5;
    if (lane == 0) {
        red[wv][0] = r.sp;  red[wv][1] = r.st;  red[wv][2] = r.spt;
        red[wv][3] = r.n1;  red[wv][4] = r.mx;  red[wv][5] = r.ce;
        red[wv][6] = r.ces;
    }
    __syncthreads();
    if (tid == 0) {
        Red b;
        b.sp = red[0][0]; b.st = red[0][1]; b.spt = red[0][2]; b.n1 = red[0][3];
        b.mx = red[0][4]; b.ce = red[0][5]; b.ces = red[0][6];
        for (int w = 1; w < TPB / 32; ++w) {
            b.sp += red[w][0]; b.st += red[w][1]; b.spt += red[w][2]; b.n1 += red[w][3];
            combine(b, red[w][4], red[w][5], red[w][6]);
        }
        float* q = part + ((size_t)row * BPR + blk) * PART_STRIDE;
        q[0] = b.sp; q[1] = b.st; q[2] = b.spt; q[3] = b.n1;
        q[4] = b.mx; q[5] = b.ce; q[6] = b.ces; q[7] = 0.f;
    }
}

__global__ __launch_bounds__(64)
void jaccard_final(const float* __restrict__ part, float* __restrict__ out) {
    __shared__ float ls[BATCH];
    const int b = threadIdx.x;
    float term = 0.f;
    if (b < BATCH) {
        Red r;
        r.sp = 0.f; r.st = 0.f; r.spt = 0.f; r.n1 = 0.f;
        r.mx = -__builtin_inff(); r.ce = 0.f; r.ces = 0.f;
        for (int k = 0; k < BPR; ++k) {
            const float* q = part + ((size_t)b * BPR + k) * PART_STRIDE;
            r.sp += q[0]; r.st += q[1]; r.spt += q[2]; r.n1 += q[3];
            combine(r, q[4], q[5], q[6]);
        }
        // any_pos == (gt_max > 0); otherwise GT == 0 everywhere
        float m  = (r.mx > 0.0f) ? r.ce  : 0.0f;
        float m1 = (r.mx > 0.0f) ? r.ces : 0.0f;
        float corr = (float)NELEM - m - r.n1 + 2.0f * m1;
        float acc  = corr / (float)NELEM;
        float inter = r.spt;
        float uni   = r.sp + r.st - inter;
        float score = (inter + 1.0f) / (uni + 1.0f);   // SMOOTH = 1.0
        if (acc == 1.0f) score = 1.0f;
        term = 1.0f - score;
    }
    ls[b] = term;
    __syncthreads();
    if (b == 0) {
        float s = 0.f;
        for (int i = 0; i < BATCH; ++i) s += ls[i];
        out[0] = s;
    }
}

extern "C" void kernel_launch(void* const* d_in, const int* in_sizes, int n_in,
                              void* d_out, int out_size, void* d_ws, size_t ws_size,
                              hipStream_t stream) {
    (void)in_sizes; (void)n_in; (void)out_size; (void)ws_size;
    const float* probs   = (const float*)d_in[0];
    const float* targets = (const float*)d_in[1];
    float* out  = (float*)d_out;
    float* part = (float*)d_ws;   // needs BATCH*BPR*PART_STRIDE*4 = 64 KB

    dim3 grid(BPR, BATCH);
    hipLaunchKernelGGL(jaccard_partial, grid, dim3(TPB), 0, stream,
                       probs, targets, part);
    hipLaunchKernelGGL(jaccard_final, dim3(1), dim3(64), 0, stream,
                       part, out);
}